// multichannel_GCN_83468394430689
// MI455X (gfx1250) — compile-verified
//
#include <hip/hip_runtime.h>

// ---------------------------------------------------------------------------
// multichannel GCN for MI455X (gfx1250, wave32)
//   - fp32 GEMMs via V_WMMA_F32_16X16X4_F32 (exact fp32 matrix path)
//   - weights pre-transposed so every A/B fragment is one global_load_b64
//   - 2 M-tiles per wave: 16 WMMAs per 10 vmem loads per k-step
//   - edge gather/scatter-add kept L2-resident (25.6MB feature arrays << 192MB L2)
//   - degree scalings fused into GEMM A-load / epilogue
// ---------------------------------------------------------------------------

typedef __attribute__((ext_vector_type(2))) float v2f;
typedef __attribute__((ext_vector_type(8))) float v8f;

// --------------------------- utility kernels -------------------------------

__global__ void zero_f4_kernel(float4* __restrict__ p, int n4) {
    int i = blockIdx.x * blockDim.x + threadIdx.x;
    if (i < n4) p[i] = make_float4(0.f, 0.f, 0.f, 0.f);
}

// outT[N x K] = in[K x N]^T   (tiny weight matrices; writes coalesced)
__global__ void transpose_kernel(const float* __restrict__ in,
                                 float* __restrict__ outT, int K, int N) {
    int i = blockIdx.x * blockDim.x + threadIdx.x;
    if (i < N * K) {
        int n = i / K, k = i % K;
        outT[i] = in[k * N + n];
    }
}

__global__ void degree_count_kernel(const int* __restrict__ src,
                                    const int* __restrict__ dst,
                                    float* __restrict__ deg_out,
                                    float* __restrict__ deg_in, int nE) {
    int e = blockIdx.x * blockDim.x + threadIdx.x;
    if (e < nE) {
        atomicAdd(&deg_out[src[e]], 1.0f);
        atomicAdd(&deg_in[dst[e]], 1.0f);
    }
}

__global__ void rsqrt_clamp_kernel(float* __restrict__ d, int n) {
    int i = blockIdx.x * blockDim.x + threadIdx.x;
    if (i < n) d[i] = rsqrtf(fmaxf(d[i], 1.0f));
}

// One warp (32 lanes) covers one edge row: lane -> 16B chunk => coalesced
// 512B gather of h[src[e]] and 128 L2-resident f32 atomic adds into agg[dst[e]].
__global__ void edge_aggregate_kernel(const int* __restrict__ src,
                                      const int* __restrict__ dst,
                                      const float* __restrict__ h,
                                      float* __restrict__ agg, int nE) {
    int tid = blockIdx.x * blockDim.x + threadIdx.x;
    int e = tid >> 5;
    int f = (tid & 31) * 4;
    if (e < nE) {
        const float4 v = *(const float4*)(h + (size_t)src[e] * 128 + f);
        float* a = agg + (size_t)dst[e] * 128 + f;
        atomicAdd(a + 0, v.x);
        atomicAdd(a + 1, v.y);
        atomicAdd(a + 2, v.z);
        atomicAdd(a + 3, v.w);
    }
}

// out_feat[ids[r]] += h[r]   (ids may repeat across/within links -> atomics)
__global__ void scatter_rows_kernel(const int* __restrict__ ids,
                                    const float* __restrict__ h,
                                    float* __restrict__ out_feat, int n) {
    int tid = blockIdx.x * blockDim.x + threadIdx.x;
    int r = tid >> 5;
    int f = (tid & 31) * 4;
    if (r < n) {
        const float4 v = *(const float4*)(h + (size_t)r * 128 + f);
        float* o = out_feat + (size_t)ids[r] * 128 + f;
        atomicAdd(o + 0, v.x);
        atomicAdd(o + 1, v.y);
        atomicAdd(o + 2, v.z);
        atomicAdd(o + 3, v.w);
    }
}

// --------------------------- WMMA fp32 GEMM --------------------------------
// out[M x N] = act( (arowscale ? arowscale[m] : 1) * A[M x 128] @ W[128 x N]
//                   + bias[N] ) * (erowscale ? erowscale[m] : 1)
// Wt is the pre-transposed weight, N x 128 row-major, so a lane's B fragment
// (Wt[n][k0], Wt[n][k0+1]) is one contiguous b64 load.
// N = NTILES*16. One wave owns TWO 16 x N output strips (B fragments reused):
//   - A fragment (v2f): lane&15 = row, (lane>>4)*2 = K pair
//   - B fragment (v2f): col = lane&15 (+16*nt), K = (lane>>4)*2 (+1)
//   - C/D (v8f): VGPR j holds rows j (lanes 0-15) and j+8 (lanes 16-31)
// EXEC stays all-1s for every WMMA (guards are wave-uniform; tile-1 rows are
// clamped for loads and its store is guarded).
template <int NTILES>
__global__ __launch_bounds__(256)
void gemm_wmma_k128_kernel(const float* __restrict__ A,
                           const float* __restrict__ Wt,   // N x 128
                           const float* __restrict__ bias, // N
                           const float* __restrict__ arowscale,
                           const float* __restrict__ erowscale,
                           float* __restrict__ out,        // M x N
                           int M, int relu) {
    const int lane    = threadIdx.x & 31;
    const int wave    = threadIdx.x >> 5;
    const int task    = blockIdx.x * (blockDim.x >> 5) + wave;
    const int t0      = task * 2;        // first 16-row tile
    const int t1      = t0 + 1;          // second 16-row tile
    if (t0 * 16 >= M) return;            // wave-uniform exit
    const bool has1   = (t1 * 16 < M);   // wave-uniform
    const int halfsel = lane >> 4;       // 0 | 1
    const int lid     = lane & 15;
    const int N       = NTILES * 16;

    v8f acc0[NTILES], acc1[NTILES];
#pragma unroll
    for (int nt = 0; nt < NTILES; ++nt) {
        float bv = bias[nt * 16 + lid];  // bias depends only on column
#pragma unroll
        for (int j = 0; j < 8; ++j) { acc0[nt][j] = bv; acc1[nt][j] = bv; }
    }

    const int arow0 = t0 * 16 + lid;
    const int arow1 = min(t1 * 16 + lid, M - 1);   // clamped: safe loads
    const float* aptr0 = A + (size_t)arow0 * 128 + halfsel * 2;
    const float* aptr1 = A + (size_t)arow1 * 128 + halfsel * 2;
    const float ascale0 = arowscale ? arowscale[arow0] : 1.0f;
    const float ascale1 = arowscale ? arowscale[arow1] : 1.0f;

    // lane-fixed base into Wt: row = lid (within an n-tile), k slot = halfsel*2
    const float* wlane = Wt + (size_t)lid * 128 + halfsel * 2;

#pragma unroll 2
    for (int kk = 0; kk < 32; ++kk) {
        v2f a0 = *(const v2f*)(aptr0);
        v2f a1 = *(const v2f*)(aptr1);
        a0.x *= ascale0; a0.y *= ascale0;
        a1.x *= ascale1; a1.y *= ascale1;
        aptr0 += 4; aptr1 += 4;
#pragma unroll
        for (int nt = 0; nt < NTILES; ++nt) {
            // Wt[(nt*16+lid)][kk*4 + halfsel*2 .. +1] : one b64 load
            v2f b = *(const v2f*)(wlane + (size_t)nt * 16 * 128 + kk * 4);
            acc0[nt] = __builtin_amdgcn_wmma_f32_16x16x4_f32(
                false, a0, false, b, (short)0, acc0[nt], false, false);
            acc1[nt] = __builtin_amdgcn_wmma_f32_16x16x4_f32(
                false, a1, false, b, (short)0, acc1[nt], false, false);
        }
    }

    // epilogue: tile 0
#pragma unroll
    for (int nt = 0; nt < NTILES; ++nt) {
#pragma unroll
        for (int j = 0; j < 8; ++j) {
            float v = acc0[nt][j];
            if (relu) v = fmaxf(v, 0.0f);
            const int row = t0 * 16 + j + halfsel * 8;
            if (erowscale) v *= erowscale[row];
            out[(size_t)row * N + nt * 16 + lid] = v;
        }
    }
    // epilogue: tile 1 (guarded, wave-uniform; M % 16 == 0 in all call sites)
    if (has1) {
#pragma unroll
        for (int nt = 0; nt < NTILES; ++nt) {
#pragma unroll
            for (int j = 0; j < 8; ++j) {
                float v = acc1[nt][j];
                if (relu) v = fmaxf(v, 0.0f);
                const int row = t1 * 16 + j + halfsel * 8;
                if (erowscale) v *= erowscale[row];
                out[(size_t)row * N + nt * 16 + lid] = v;
            }
        }
    }
}

// ------------------------------ launcher -----------------------------------

extern "C" void kernel_launch(void* const* d_in, const int* in_sizes, int n_in,
                              void* d_out, int out_size, void* d_ws, size_t ws_size,
                              hipStream_t stream) {
    (void)n_in; (void)ws_size;
    const float* x   = (const float*)d_in[0];   // [L, NS, H]
    const float* W2  = (const float*)d_in[1];   // [L, H, H]
    const float* b2  = (const float*)d_in[2];   // [L, H]
    const float* gw  = (const float*)d_in[3];   // [L, NL, H, H]
    const float* gb  = (const float*)d_in[4];   // [L, NL, H]
    const float* Wc  = (const float*)d_in[5];   // [H, C]
    const float* bc  = (const float*)d_in[6];   // [C]
    const int*   src = (const int*)d_in[7];     // [L, NE]
    const int*   dst = (const int*)d_in[8];     // [L, NE]
    const int*   ids = (const int*)d_in[9];     // [L, NS]

    const int H  = 128;
    const int C  = 16;
    const int L  = in_sizes[2] / H;             // 3
    const int NS = in_sizes[9] / L;             // 50000
    const int NE = in_sizes[7] / L;             // 800000
    const int NL = in_sizes[4] / (L * H);       // 2
    const int NN = out_size / C;                // 100000 (num_nodes)

    // workspace layout (floats)
    float* ws       = (float*)d_ws;
    float* out_feat = ws;                                   // NN*H
    float* hbuf     = out_feat + (size_t)NN * H;            // NS*H
    float* agg      = hbuf + (size_t)NS * H;                // NS*H
    float* deg_out  = agg + (size_t)NS * H;                 // NS
    float* deg_in   = deg_out + NS;                         // NS
    float* w2t      = deg_in + NS;                          // L*H*H
    float* gwt      = w2t + (size_t)L * H * H;              // L*NL*H*H
    float* wct      = gwt + (size_t)L * NL * H * H;         // C*H

    const int T = 256;
    const int feat_blocks_NS = (NS * 32 + T - 1) / T;       // warp-per-row kernels
    // 2 M-tiles per wave, 8 waves per block
    const int gemm_blocks_NS = (((NS / 16) + 1) / 2 + 7) / 8;
    const int gemm_blocks_NN = (((NN / 16) + 1) / 2 + 7) / 8;

    // ---- pre-transpose all weight matrices (tiny; amortized over GEMMs) ----
    {
        int nHH = H * H;                           // 16384
        int bHH = (nHH + T - 1) / T;
        for (int i = 0; i < L; ++i)
            transpose_kernel<<<bHH, T, 0, stream>>>(W2 + (size_t)i * nHH,
                                                    w2t + (size_t)i * nHH, H, H);
        for (int i = 0; i < L * NL; ++i)
            transpose_kernel<<<bHH, T, 0, stream>>>(gw + (size_t)i * nHH,
                                                    gwt + (size_t)i * nHH, H, H);
        int nHC = H * C;
        transpose_kernel<<<(nHC + T - 1) / T, T, 0, stream>>>(Wc, wct, H, C);
    }

    // out_feat = 0
    {
        int n4 = (int)((size_t)NN * H / 4);
        zero_f4_kernel<<<(n4 + T - 1) / T, T, 0, stream>>>((float4*)out_feat, n4);
    }

    for (int i = 0; i < L; ++i) {
        const int*   src_i = src + (size_t)i * NE;
        const int*   dst_i = dst + (size_t)i * NE;
        const int*   ids_i = ids + (size_t)i * NS;
        const float* x_i   = x   + (size_t)i * NS * H;
        const float* w2t_i = w2t + (size_t)i * H * H;
        const float* b2_i  = b2  + (size_t)i * H;

        // degrees (shared by both conv layers of this link)
        {
            int n4 = (2 * NS) / 4;
            zero_f4_kernel<<<(n4 + T - 1) / T, T, 0, stream>>>((float4*)deg_out, n4);
        }
        degree_count_kernel<<<(NE + T - 1) / T, T, 0, stream>>>(src_i, dst_i,
                                                                deg_out, deg_in, NE);
        rsqrt_clamp_kernel<<<(2 * NS + T - 1) / T, T, 0, stream>>>(deg_out, 2 * NS);

        // feature2: h = x@W2 + b2, pre-scaled by rsqrt(deg_out) for the gather
        gemm_wmma_k128_kernel<8><<<gemm_blocks_NS, T, 0, stream>>>(
            x_i, w2t_i, b2_i, nullptr, deg_out, hbuf, NS, 0);

        for (int l = 0; l < NL; ++l) {
            const float* gwt_il = gwt + ((size_t)i * NL + l) * H * H;
            const float* gb_il  = gb  + ((size_t)i * NL + l) * H;
            {
                int n4 = (int)((size_t)NS * H / 4);
                zero_f4_kernel<<<(n4 + T - 1) / T, T, 0, stream>>>((float4*)agg, n4);
            }
            edge_aggregate_kernel<<<(NE * 32) / T, T, 0, stream>>>(src_i, dst_i,
                                                                   hbuf, agg, NE);
            // h = relu( (agg * rsqrt(deg_in)) @ gw + gb )
            //     * (rsqrt(deg_out) if another conv layer follows)
            const float* escale = (l + 1 < NL) ? deg_out : nullptr;
            gemm_wmma_k128_kernel<8><<<gemm_blocks_NS, T, 0, stream>>>(
                agg, gwt_il, gb_il, deg_in, escale, hbuf, NS, 1);
        }

        // out_feat[ids[i]] += h
        scatter_rows_kernel<<<feat_blocks_NS, T, 0, stream>>>(ids_i, hbuf,
                                                              out_feat, NS);
    }

    // classifier: d_out = out_feat @ Wc + bc   (N = 16 -> one N tile)
    gemm_wmma_k128_kernel<1><<<gemm_blocks_NN, T, 0, stream>>>(
        out_feat, wct, bc, nullptr, nullptr, (float*)d_out, NN, 0);
}